// SetAbstractionLevel_61418032333282
// MI455X (gfx1250) — compile-verified
//
#include <hip/hip_runtime.h>

#define B_   16
#define N_   4096
#define S_   1024
#define K_   32
#define D_   64
#define C1   67
#define C1P  68
#define O1   64
#define O2   64
#define O3   128
#define M_   (B_ * S_ * K_)   /* 524288 */
#define RAD2 0.04f
#define EPS_ 1e-5f

typedef __attribute__((ext_vector_type(2))) float v2f;
typedef __attribute__((ext_vector_type(8))) float v8f;

// ---------------------------------------------------------------------------
// 1) Farthest point sampling: one block per batch, xyz + dist in LDS.
// ---------------------------------------------------------------------------
__global__ __launch_bounds__(256) void fps_kernel(const float* __restrict__ xyz,
                                                  float* __restrict__ nxyz,
                                                  float* __restrict__ out_xyz) {
    const int b   = blockIdx.x;
    const int tid = threadIdx.x;
    __shared__ float lx[N_], ly[N_], lz[N_], dist[N_];
    __shared__ float rv[8];
    __shared__ int   ri[8];
    __shared__ int   sFar;

    const float* xb = xyz + (size_t)b * 3 * N_;
    for (int n = tid; n < N_; n += 256) {
        lx[n] = xb[n]; ly[n] = xb[N_ + n]; lz[n] = xb[2 * N_ + n];
        dist[n] = 1e10f;
    }
    if (tid == 0) sFar = 0;
    __syncthreads();

    for (int i = 0; i < S_; ++i) {
        const int far = sFar;
        const float fx = lx[far], fy = ly[far], fz = lz[far];
        if (tid == 0) {
            nxyz[((size_t)b * S_ + i) * 3 + 0] = fx;
            nxyz[((size_t)b * S_ + i) * 3 + 1] = fy;
            nxyz[((size_t)b * S_ + i) * 3 + 2] = fz;
            out_xyz[(size_t)b * 3 * S_ + 0 * S_ + i] = fx;
            out_xyz[(size_t)b * 3 * S_ + 1 * S_ + i] = fy;
            out_xyz[(size_t)b * 3 * S_ + 2 * S_ + i] = fz;
        }
        float bv = -1.0f; int bi = N_;
        for (int n = tid; n < N_; n += 256) {
            const float dx = lx[n] - fx, dy = ly[n] - fy, dz = lz[n] - fz;
            const float d  = dx * dx + dy * dy + dz * dz;
            const float nd = fminf(dist[n], d);
            dist[n] = nd;
            if (nd > bv || (nd == bv && n < bi)) { bv = nd; bi = n; }
        }
        // wave32 reduce (max value, min index on tie == jnp.argmax semantics)
        for (int off = 16; off > 0; off >>= 1) {
            const float ov = __shfl_down(bv, off);
            const int   oi = __shfl_down(bi, off);
            if (ov > bv || (ov == bv && oi < bi)) { bv = ov; bi = oi; }
        }
        if ((tid & 31) == 0) { rv[tid >> 5] = bv; ri[tid >> 5] = bi; }
        __syncthreads();
        if (tid == 0) {
            float v = rv[0]; int ix = ri[0];
            for (int w = 1; w < 8; ++w)
                if (rv[w] > v || (rv[w] == v && ri[w] < ix)) { v = rv[w]; ix = ri[w]; }
            sFar = ix;
        }
        __syncthreads();
    }
}

// ---------------------------------------------------------------------------
// 2) Ball query: thread per query, xyz staged in LDS, first-32 in index order.
// ---------------------------------------------------------------------------
__global__ __launch_bounds__(256) void ballq_kernel(const float* __restrict__ xyz,
                                                    const float* __restrict__ nxyz,
                                                    int* __restrict__ gidx) {
    const int blk = blockIdx.x;            // 64 blocks
    const int b   = blk >> 2;
    const int s   = ((blk & 3) << 8) + threadIdx.x;
    __shared__ float lx[N_], ly[N_], lz[N_];
    const float* xb = xyz + (size_t)b * 3 * N_;
    for (int n = threadIdx.x; n < N_; n += 256) {
        lx[n] = xb[n]; ly[n] = xb[N_ + n]; lz[n] = xb[2 * N_ + n];
    }
    __syncthreads();

    const size_t bs = (size_t)b * S_ + s;
    const float qx = nxyz[bs * 3 + 0], qy = nxyz[bs * 3 + 1], qz = nxyz[bs * 3 + 2];
    int* gp = gidx + bs * K_;
    int cnt = 0, first = 0;
    for (int n = 0; n < N_; ++n) {
        const float dx = lx[n] - qx, dy = ly[n] - qy, dz = lz[n] - qz;
        const float d = dx * dx + dy * dy + dz * dz;
        if (d <= RAD2) {                   // keep if NOT (sqr > r^2)
            if (cnt == 0) first = n;
            gp[cnt++] = n;
            if (cnt == K_) break;
        }
    }
    for (int q = cnt; q < K_; ++q) gp[q] = first;
}

// ---------------------------------------------------------------------------
// 3) Gather + concat -> X1 rows [M, 68] (ch 0-2 rel-xyz, 3-66 feats, 67 zero).
// ---------------------------------------------------------------------------
__global__ __launch_bounds__(128) void gather_kernel(const float* __restrict__ xyz,
                                                     const float* __restrict__ points,
                                                     const float* __restrict__ nxyz,
                                                     const int* __restrict__ gidx,
                                                     float* __restrict__ X1) {
    const int wid  = threadIdx.x >> 5;
    const int lane = threadIdx.x & 31;
    const size_t m = (size_t)blockIdx.x * 4 + wid;
    const int bs = (int)(m >> 5);          // query index
    const int b  = bs >> 10;               // / S_
    const int j  = gidx[m];
    float* xr = X1 + m * C1P;
    for (int c = lane; c < C1P; c += 32) {
        float v;
        if (c < 3)       v = xyz[((size_t)b * 3 + c) * N_ + j] - nxyz[(size_t)bs * 3 + c];
        else if (c < C1) v = points[((size_t)b * D_ + (c - 3)) * N_ + j];
        else             v = 0.0f;
        xr[c] = v;
    }
}

// ---------------------------------------------------------------------------
// 4) Fused GEMM (+optional input affine+ReLU) + bias + BN-stat accumulation.
//    D = A x B with V_WMMA_F32_16X16X4_F32; each block owns 128 M-rows and
//    ALL output tiles (full W in LDS) so X streams from HBM exactly once.
// ---------------------------------------------------------------------------
template <bool APPLY, int NOT>
__global__ __launch_bounds__(256) void gemm_bn_kernel(const float* __restrict__ X, int Cpad,
                                                      const float* __restrict__ W, int Creal,
                                                      const float* __restrict__ bias,
                                                      const float* __restrict__ inPar,
                                                      float* __restrict__ Y,
                                                      float* __restrict__ stats) {
    const int O = NOT * 16;
    __shared__ float Wlds[O3 * C1P];       // 34.8 KB worst case
    __shared__ float aArr[C1P], cArr[C1P];
    __shared__ float bSum[O3], bSumSq[O3];
    const int tid = threadIdx.x;

    for (int t = tid; t < O * Cpad; t += 256) {
        const int rr = t / Cpad, cc = t - rr * Cpad;
        Wlds[t] = (cc < Creal) ? W[(size_t)rr * Creal + cc] : 0.0f;
    }
    if (APPLY)
        for (int t = tid; t < Cpad; t += 256) { aArr[t] = inPar[t]; cArr[t] = inPar[Cpad + t]; }
    if (tid < O) { bSum[tid] = 0.0f; bSumSq[tid] = 0.0f; }
    __syncthreads();

    const int wid = tid >> 5, lane = tid & 31;
    const int h = lane >> 4, r = lane & 15;
    const int m0 = (blockIdx.x * 8 + wid) * 16;
    const float* xrow = X + (size_t)(m0 + r) * Cpad;

    v8f zero = {};
    v8f acc[NOT];
#pragma unroll
    for (int ot = 0; ot < NOT; ++ot) acc[ot] = zero;

    for (int cb = 0; cb < Cpad; cb += 4) {
        const int c0 = cb + 2 * h;         // A: lanes 0-15 K=0,1; lanes 16-31 K=2,3
        v2f av = *(const v2f*)(xrow + c0);
        if (APPLY) {
            av.x = fmaxf(av.x * aArr[c0]     + cArr[c0],     0.0f);
            av.y = fmaxf(av.y * aArr[c0 + 1] + cArr[c0 + 1], 0.0f);
        }
#pragma unroll
        for (int ot = 0; ot < NOT; ++ot) {
            v2f bv = *(const v2f*)(&Wlds[(size_t)(ot * 16 + r) * Cpad + c0]);
            acc[ot] = __builtin_amdgcn_wmma_f32_16x16x4_f32(
                false, av, false, bv, (short)0, acc[ot], false, false);
        }
    }

#pragma unroll
    for (int ot = 0; ot < NOT; ++ot) {
        const int o = ot * 16 + r;
        const float bo = bias[o];
        float s1 = 0.0f, s2 = 0.0f;
        float* yb = Y + o;
#pragma unroll
        for (int i = 0; i < 8; ++i) {      // D VGPR i -> row m0+i+8h, col o
            const float y = acc[ot][i] + bo;
            yb[(size_t)(m0 + i + 8 * h) * O] = y;
            s1 += y; s2 += y * y;
        }
        s1 += __shfl_xor(s1, 16);
        s2 += __shfl_xor(s2, 16);
        if (lane < 16) { atomicAdd(&bSum[o], s1); atomicAdd(&bSumSq[o], s2); }
    }
    __syncthreads();
    if (tid < O) {
        atomicAdd(&stats[tid], bSum[tid]);
        atomicAdd(&stats[128 + tid], bSumSq[tid]);
    }
}

// ---------------------------------------------------------------------------
// 5) Fold BN stats into affine params: a = g*rsqrt(var+eps), c = be - mean*a.
// ---------------------------------------------------------------------------
__global__ void finalize_kernel(const float* __restrict__ stats, const float* __restrict__ g,
                                const float* __restrict__ be, float* __restrict__ par, int O) {
    const int o = threadIdx.x;
    if (o < O) {
        const float invM = 1.0f / (float)M_;
        const float mean = stats[o] * invM;
        const float var  = stats[128 + o] * invM - mean * mean;
        const float a = g[o] * rsqrtf(var + EPS_);
        par[o]     = a;
        par[O + o] = be[o] - mean * a;
    }
}

// ---------------------------------------------------------------------------
// 6) Final affine + ReLU + max over K=32 samples -> (B,128,S) output.
// ---------------------------------------------------------------------------
__global__ __launch_bounds__(128) void pool_kernel(const float* __restrict__ Y3,
                                                   const float* __restrict__ par,
                                                   float* __restrict__ out) {
    const int bs = blockIdx.x;             // B*S
    const int o  = threadIdx.x;            // 128
    const float a = par[o], c2 = par[O3 + o];
    const float* yb = Y3 + (size_t)bs * K_ * O3 + o;
    float mx = 0.0f;
#pragma unroll 4
    for (int k = 0; k < K_; ++k) {
        const float v = fmaxf(a * yb[(size_t)k * O3] + c2, 0.0f);
        mx = fmaxf(mx, v);
    }
    const int b = bs >> 10, s = bs & (S_ - 1);
    out[(size_t)b * O3 * S_ + (size_t)o * S_ + s] = mx;
}

// ---------------------------------------------------------------------------
extern "C" void kernel_launch(void* const* d_in, const int* in_sizes, int n_in,
                              void* d_out, int out_size, void* d_ws, size_t ws_size,
                              hipStream_t stream) {
    (void)in_sizes; (void)n_in; (void)out_size; (void)ws_size;
    const float* xyz    = (const float*)d_in[0];
    const float* points = (const float*)d_in[1];
    const float* W1 = (const float*)d_in[2];
    const float* b1 = (const float*)d_in[3];
    const float* g1 = (const float*)d_in[4];
    const float* be1 = (const float*)d_in[5];
    const float* W2 = (const float*)d_in[6];
    const float* b2 = (const float*)d_in[7];
    const float* g2 = (const float*)d_in[8];
    const float* be2 = (const float*)d_in[9];
    const float* W3 = (const float*)d_in[10];
    const float* b3 = (const float*)d_in[11];
    const float* g3 = (const float*)d_in[12];
    const float* be3 = (const float*)d_in[13];
    float* out = (float*)d_out;

    char* ws = (char*)d_ws;
    size_t off = 0;
    auto alloc = [&](size_t bytes) -> void* {
        void* p = ws + off;
        off = (off + bytes + 255) & ~(size_t)255;
        return p;
    };
    float* nxyz  = (float*)alloc((size_t)B_ * S_ * 3 * sizeof(float));
    int*   gidx  = (int*)  alloc((size_t)B_ * S_ * K_ * sizeof(int));
    float* stats = (float*)alloc(3 * 256 * sizeof(float));
    float* par1  = (float*)alloc(256 * sizeof(float));
    float* par2  = (float*)alloc(256 * sizeof(float));
    float* par3  = (float*)alloc(256 * sizeof(float));
    float* X1 = (float*)alloc((size_t)M_ * C1P * sizeof(float));
    float* Y1 = (float*)alloc((size_t)M_ * O1 * sizeof(float));
    float* Y2 = (float*)alloc((size_t)M_ * O2 * sizeof(float));
    float* Y3 = (float*)alloc((size_t)M_ * O3 * sizeof(float));

    hipMemsetAsync(stats, 0, 3 * 256 * sizeof(float), stream);

    fps_kernel<<<B_, 256, 0, stream>>>(xyz, nxyz, out);
    ballq_kernel<<<64, 256, 0, stream>>>(xyz, nxyz, gidx);
    gather_kernel<<<M_ / 4, 128, 0, stream>>>(xyz, points, nxyz, gidx, X1);

    gemm_bn_kernel<false, O1 / 16><<<M_ / 128, 256, 0, stream>>>(
        X1, C1P, W1, C1, b1, nullptr, Y1, stats);
    finalize_kernel<<<1, 128, 0, stream>>>(stats, g1, be1, par1, O1);

    gemm_bn_kernel<true, O2 / 16><<<M_ / 128, 256, 0, stream>>>(
        Y1, O1, W2, O1, b2, par1, Y2, stats + 256);
    finalize_kernel<<<1, 128, 0, stream>>>(stats + 256, g2, be2, par2, O2);

    gemm_bn_kernel<true, O3 / 16><<<M_ / 128, 256, 0, stream>>>(
        Y2, O2, W3, O2, b3, par2, Y3, stats + 512);
    finalize_kernel<<<1, 128, 0, stream>>>(stats + 512, g3, be3, par3, O3);

    pool_kernel<<<B_ * S_, 128, 0, stream>>>(Y3, par3, out + (size_t)B_ * 3 * S_);
}